// KernalAnsatz_65481071405325
// MI455X (gfx1250) — compile-verified
//
#include <hip/hip_runtime.h>
#include <hip/hip_bf16.h>
#include <math.h>

typedef __attribute__((ext_vector_type(2))) float v2f;
typedef __attribute__((ext_vector_type(4))) float v4f;
typedef __attribute__((ext_vector_type(8))) float v8f;

#define D_DIM 128              // feature dimension
#define M_TILE 128             // output rows per block
#define N_TILE 64              // output cols per block
#define LDS_STRIDE 132         // 128 + 4 pad floats -> conflict-free DS_LOAD_B64

// ---------------------------------------------------------------------------
// Kernel 1: row norms.  One wave32 per row; 8 rows per 256-thread block.
// blockIdx.y == 0 -> x rows, == 1 -> y rows.  Output packed in d_ws.
// ---------------------------------------------------------------------------
__global__ __launch_bounds__(256) void rbf_norms_kernel(
    const float* __restrict__ x, const float* __restrict__ y,
    float* __restrict__ ws, int nx, int ny) {
  const int sel = blockIdx.y;
  const float* src = sel ? y : x;
  const int nrows = sel ? ny : nx;
  float* dst = ws + (sel ? nx : 0);

  const int wave = threadIdx.x >> 5;
  const int lane = threadIdx.x & 31;
  const int row = blockIdx.x * 8 + wave;
  if (row >= nrows) return;

  const float* p = src + (size_t)row * D_DIM;
  float s = 0.0f;
#pragma unroll
  for (int i = 0; i < D_DIM; i += 32) {
    float v = p[i + lane];
    s = fmaf(v, v, s);
  }
  // wave32 butterfly reduction
#pragma unroll
  for (int m = 16; m >= 1; m >>= 1) s += __shfl_xor(s, m, 32);
  if (lane == 0) dst[row] = s;
}

// ---------------------------------------------------------------------------
// Kernel 2: fused GEMM (V_WMMA_F32_16X16X4_F32) + RBF epilogue.
// Block tile: M_TILE x N_TILE.  8 waves, each wave -> 16 x N_TILE strip.
// ---------------------------------------------------------------------------
__global__ __launch_bounds__(256) void rbf_wmma_kernel(
    const float* __restrict__ x, const float* __restrict__ y,
    const float* __restrict__ x2, const float* __restrict__ y2,
    float* __restrict__ out, int M, int N) {
  __shared__ __align__(16) float ylds[N_TILE * LDS_STRIDE];

  const int n0 = blockIdx.x * N_TILE;   // fast-varying -> coalesced row writes
  const int m0 = blockIdx.y * M_TILE;
  const int t = threadIdx.x;

  // --- cooperative stage of the y tile (64 rows x 128 f32) into LDS --------
  // 2048 float4 total, 256 threads -> 8 iterations of B128 load/store.
#pragma unroll
  for (int it = 0; it < (N_TILE * (D_DIM / 4)) / 256; ++it) {
    int f = it * 256 + t;
    int r = f >> 5;          // 32 float4 per row
    int c4 = f & 31;
    v4f v = *(const v4f*)(y + (size_t)(n0 + r) * D_DIM + c4 * 4);
    *(v4f*)(&ylds[r * LDS_STRIDE + c4 * 4]) = v;
  }
  __syncthreads();

  const int wave = t >> 5;
  const int lane = t & 31;
  const int lane15 = lane & 15;
  const int hi = lane >> 4;            // which half of the wave

  // --- preload A panel (16 x 128) in WMMA 16x4 f32 layout ------------------
  // lane holds row M = lane15; VGPR pair s holds K = 4s + 2*hi (+0,+1).
  const int mrow = m0 + wave * 16 + lane15;
  const float* xrow = x + (size_t)mrow * D_DIM + 2 * hi;
  v2f areg[D_DIM / 4];
#pragma unroll
  for (int s = 0; s < D_DIM / 4; ++s) areg[s] = *(const v2f*)(xrow + 4 * s);

  v8f acc[N_TILE / 16];
#pragma unroll
  for (int j = 0; j < N_TILE / 16; ++j) acc[j] = (v8f)0.0f;

  // B operand: lane holds col N = lane15 of subtile j; same K split as A.
  const float* bbase = &ylds[lane15 * LDS_STRIDE + 2 * hi];

#pragma unroll
  for (int s = 0; s < D_DIM / 4; ++s) {
#pragma unroll
    for (int j = 0; j < N_TILE / 16; ++j) {
      v2f b = *(const v2f*)(bbase + j * 16 * LDS_STRIDE + 4 * s);
      acc[j] = __builtin_amdgcn_wmma_f32_16x16x4_f32(
          /*neg_a=*/false, areg[s], /*neg_b=*/false, b,
          /*c_mod=*/(short)0, acc[j], /*reuse_a=*/false, /*reuse_b=*/false);
    }
  }

  // --- epilogue: sq = x2 + y2 - 2*xy, clamp, exp ---------------------------
  // C/D layout: lane L, VGPR v -> (M = v + 8*hi, N = lane15) in 16x16 tile.
  float xn[8];
  {
    const float* p = x2 + m0 + wave * 16 + 8 * hi;
    v4f a = *(const v4f*)p;
    v4f bq = *(const v4f*)(p + 4);
    xn[0] = a.x; xn[1] = a.y; xn[2] = a.z; xn[3] = a.w;
    xn[4] = bq.x; xn[5] = bq.y; xn[6] = bq.z; xn[7] = bq.w;
  }
  const int rbase = m0 + wave * 16 + 8 * hi;
#pragma unroll
  for (int j = 0; j < N_TILE / 16; ++j) {
    const int col = n0 + j * 16 + lane15;
    const float yn = y2[col];
#pragma unroll
    for (int v = 0; v < 8; ++v) {
      float sq = xn[v] + yn - 2.0f * acc[j][v];
      sq = fmaxf(sq, 0.0f);                 // guard cancellation negatives
      out[(size_t)(rbase + v) * N + col] = __expf(-sq);
    }
  }
}

// ---------------------------------------------------------------------------
extern "C" void kernel_launch(void* const* d_in, const int* in_sizes, int n_in,
                              void* d_out, int out_size, void* d_ws, size_t ws_size,
                              hipStream_t stream) {
  const float* x = (const float*)d_in[0];
  const float* y = (const float*)d_in[1];
  float* out = (float*)d_out;
  const int Mrows = in_sizes[0] / D_DIM;   // 8192 rows of x
  const int Nrows = in_sizes[1] / D_DIM;   // 8192 rows of y

  float* norms = (float*)d_ws;             // [Mrows] x2 followed by [Nrows] y2
  const float* x2 = norms;
  const float* y2 = norms + Mrows;

  {
    int maxr = Mrows > Nrows ? Mrows : Nrows;
    dim3 grid((maxr + 7) / 8, 2);
    rbf_norms_kernel<<<grid, 256, 0, stream>>>(x, y, norms, Mrows, Nrows);
  }
  {
    dim3 grid(Nrows / N_TILE, Mrows / M_TILE);
    rbf_wmma_kernel<<<grid, 256, 0, stream>>>(x, y, x2, y2, out, Mrows, Nrows);
  }
}